// RetrocausalAttention_56650618634408
// MI455X (gfx1250) — compile-verified
//
#include <hip/hip_runtime.h>
#include <math.h>

// ---------------------------------------------------------------------------
// Retrocausal attention (EMA forward scan + 7-tap FIR) on MI455X (gfx1250).
//
// Memory-bound (256 MB traffic ~ 11us at 23.3 TB/s). Strategy:
//  * Split T into segments with a 96-row halo: (1-alpha)^96 ~ 1e-40 == 0 in
//    fp32, so restarted EMAs are exact -> 4096 independent wave32 tiles.
//  * Per 16-row block, the EMA unrolls to a constant lower-triangular 16x16
//    matrix L (plus a rank-1 carry term), the FIR to banded matrices
//    W1 (prev block) / W2 (cur block). Output block:
//        O = (beta*W1) x Xprev + (alpha*L + beta*W2) x Xb + alpha*dvec(x)carry
//    via chains of V_WMMA_F32_16X16X4_F32 (K=16 -> 4 wmma per matmul).
//  * Carry (fwd row 15) broadcast across the wave via ds_bpermute.
//  * All loads/stores use one per-lane base pointer + compile-time immediate
//    offsets (row stride 16KB fits the 24-bit IOFFSET), advanced by a single
//    64-bit add per block -> loads clause as s_clause 0x7, minimal VALU.
// ---------------------------------------------------------------------------

typedef __attribute__((ext_vector_type(2))) float v2f;
typedef __attribute__((ext_vector_type(8))) float v8f;

#define TDIM   8192
#define DDIM   4096
#define LAG    6
#define SEG    512
#define HALO   96            // (1-alpha)^96 underflows fp32 -> exact restart
#define NSEG   (TDIM / SEG)  // 16
#define NCT    (DDIM / 16)   // 256 column tiles
#define NHALO  (HALO / 16)   // 6 warm-up blocks
#define NOUT   (SEG / 16)    // 32 output blocks
#define WAVES_PER_BLOCK 8

#define ALPHA_F   0.6180339887498949f    // 1/phi
#define BETA_F    0.3819660112501051f    // 1/phi^2 == 1 - alpha
#define OMA_F     0.3819660112501051f    // (1 - alpha)
#define LOG2_OMA  (-1.3884839899060413f) // log2(1-alpha)

// (1-alpha)^n without dynamic array indexing (register-only init math)
__device__ __forceinline__ float pw(int n) {
    return exp2f((float)n * LOG2_OMA);
}
// FIR weight w[j], j in [0, LAG]
__device__ __forceinline__ float fir_w(int j) {
    return (j < LAG) ? ALPHA_F * pw(j) : pw(LAG);
}

__global__ __launch_bounds__(32 * WAVES_PER_BLOCK)
void ema_fir_wmma(const float* __restrict__ x, float* __restrict__ out) {
    const int lane = threadIdx.x & 31;
    const int wave = threadIdx.x >> 5;
    const int tile = blockIdx.x * WAVES_PER_BLOCK + wave;
    const int colTile = tile & (NCT - 1);   // adjacent waves -> adjacent cols
    const int seg     = tile / NCT;
    const int d0 = colTile * 16;
    const int t0 = seg * SEG;

    const int col = lane & 15;   // column within tile (B/C/D layout) == A row
    const int hi  = lane >> 4;   // lane half selects K / M ranges

    // ---- Build per-lane A-matrix registers (compile-time-shaped constants).
    // A layout (f32 16x4): lanes 0-15 hold M=lane, VGPR0=K0,VGPR1=K1;
    //                      lanes 16-31 hold M=lane-16, VGPR0=K2,VGPR1=K3.
    const int m = col;
    v2f AL[4], A1[4], A2[4];
#pragma unroll
    for (int c = 0; c < 4; ++c) {
        const int kk = 4 * c + 2 * hi;
#pragma unroll
        for (int u = 0; u < 2; ++u) {
            const int k = kk + u;
            const float Lv = (k <= m) ? ALPHA_F * pw(m - k) : 0.f;
            const int j2 = k - m + LAG;        // W2 band: rows i-6..i, cur blk
            const float W2v = (j2 >= 0 && j2 <= LAG) ? fir_w(j2) : 0.f;
            const int j1 = k - m - (16 - LAG); // W1 band: tail of prev block
            const float W1v = (j1 >= 0 && j1 <= LAG) ? fir_w(j1) : 0.f;
            AL[c][u] = Lv;
            A2[c][u] = ALPHA_F * Lv + BETA_F * W2v;
            A1[c][u] = BETA_F * W1v;
        }
    }
    // Rank-1 carry coefficients for this lane's C/D rows: M = v + 8*hi.
    float adv[8];
#pragma unroll
    for (int v = 0; v < 8; ++v) adv[v] = ALPHA_F * pw(v + 8 * hi + 1);
    const float dv7 = pw(8 * hi + 8);   // (1-a)^16 on hi==1 lanes (carry decay)

    // ---- Segment setup: halo warm-up for seg>0, exact seed for seg 0.
    const int tstart = (seg == 0) ? 0 : (t0 - HALO);

    const size_t basecol   = (size_t)d0 + col;
    const size_t row16     = (size_t)16 * DDIM;        // block stride (floats)

    // Per-lane streaming pointers (immediate offsets do the rest):
    //  loads : xp + {c*4*DDIM, c*4*DDIM + DDIM}  -> rows 4c+2hi, 4c+2hi+1
    //  stores: op + v*DDIM                        -> rows 8hi + v
    const float* xp = x + (size_t)(tstart + 2 * hi) * DDIM + basecol;
    float*       op = out + (size_t)(t0 + 8 * hi) * DDIM + basecol;

    // carry seed: fwd[tstart-1] := x[tstart] makes fwd[tstart] = x[tstart]
    float carry = x[(size_t)tstart * DDIM + basecol];

    // Xprev (B layout: V[c] -> rows {4c+2*hi, +1}, column = lane&15)
    v2f P[4];
    if (seg == 0) {
#pragma unroll
        for (int c = 0; c < 4; ++c) { P[c][0] = 0.f; P[c][1] = 0.f; }
    } else {
        const float* pp = xp - row16;
#pragma unroll
        for (int c = 0; c < 4; ++c) {
            P[c][0] = pp[(size_t)c * 4 * DDIM];
            P[c][1] = pp[(size_t)c * 4 * DDIM + DDIM];
        }
    }

    // ---- Phase 1: halo warm-up (EMA carry only; no FIR, no stores).
    if (seg != 0) {
        for (int b = 0; b < NHALO; ++b) {
            v2f Bv[4];
#pragma unroll
            for (int c = 0; c < 4; ++c) {
                Bv[c][0] = xp[(size_t)c * 4 * DDIM];
                Bv[c][1] = xp[(size_t)c * 4 * DDIM + DDIM];
            }
            v8f F = {};
#pragma unroll
            for (int c = 0; c < 4; ++c)
                F = __builtin_amdgcn_wmma_f32_16x16x4_f32(
                        false, AL[c], false, Bv[c], (short)0, F, false, false);
            const float f15 = F[7] + dv7 * carry;   // row 15, hi==1 lanes
            carry = __builtin_bit_cast(
                float, __builtin_amdgcn_ds_bpermute(
                           (col + 16) << 2, __builtin_bit_cast(int, f15)));
#pragma unroll
            for (int c = 0; c < 4; ++c) P[c] = Bv[c];
            xp += row16;
        }
    }

    // ---- Phase 2: output blocks.
    for (int b = 0; b < NOUT; ++b) {
        if (b < NOUT - 2)   // uniform branch; speculative stream prefetch
            __builtin_prefetch(xp + 2 * row16, 0, 0);

        v2f Bv[4];
#pragma unroll
        for (int c = 0; c < 4; ++c) {
            Bv[c][0] = xp[(size_t)c * 4 * DDIM];
            Bv[c][1] = xp[(size_t)c * 4 * DDIM + DDIM];
        }

        // F = L x Xb  (EMA local part; row 15 feeds the carry chain)
        v8f F = {};
#pragma unroll
        for (int c = 0; c < 4; ++c)
            F = __builtin_amdgcn_wmma_f32_16x16x4_f32(
                    false, AL[c], false, Bv[c], (short)0, F, false, false);

        // O = beta*W1 x Xprev + (alpha*L + beta*W2) x Xb
        v8f O = {};
#pragma unroll
        for (int c = 0; c < 4; ++c)
            O = __builtin_amdgcn_wmma_f32_16x16x4_f32(
                    false, A1[c], false, P[c], (short)0, O, false, false);
#pragma unroll
        for (int c = 0; c < 4; ++c)
            O = __builtin_amdgcn_wmma_f32_16x16x4_f32(
                    false, A2[c], false, Bv[c], (short)0, O, false, false);

        // + alpha * (1-a)^{M+1} * carry  (rank-1 EMA carry-in term)
#pragma unroll
        for (int v = 0; v < 8; ++v) O[v] += adv[v] * carry;

        // Store: lane writes rows M = v + 8*hi, col = lane&15 (imm offsets).
#pragma unroll
        for (int v = 0; v < 8; ++v)
            op[(size_t)v * DDIM] = O[v];

        // Carry update: fwd row 15 lives in VGPR7 of hi==1 lanes.
        const float f15 = F[7] + dv7 * carry;
        carry = __builtin_bit_cast(
            float, __builtin_amdgcn_ds_bpermute(
                       (col + 16) << 2, __builtin_bit_cast(int, f15)));

#pragma unroll
        for (int c = 0; c < 4; ++c) P[c] = Bv[c];
        xp += row16;
        op += row16;
    }
}

// Exact reference semantics for t < LAG: bwd[t] = x[t], fwd by direct scan.
__global__ void fix_head(const float* __restrict__ x, float* __restrict__ out) {
    const int d = blockIdx.x * blockDim.x + threadIdx.x;
    if (d >= DDIM) return;
    float f = x[d];                       // fwd[0] = x[0]
    out[d] = ALPHA_F * f + BETA_F * x[d];
#pragma unroll
    for (int t = 1; t < LAG; ++t) {
        const float xv = x[(size_t)t * DDIM + d];
        f = ALPHA_F * xv + OMA_F * f;
        out[(size_t)t * DDIM + d] = ALPHA_F * f + BETA_F * xv;
    }
}

extern "C" void kernel_launch(void* const* d_in, const int* in_sizes, int n_in,
                              void* d_out, int out_size, void* d_ws, size_t ws_size,
                              hipStream_t stream) {
    const float* x = (const float*)d_in[0];
    float* out = (float*)d_out;

    const int tiles = NSEG * NCT;                    // 4096 wave tiles
    const int grid  = tiles / WAVES_PER_BLOCK;       // 512 blocks x 256 thr
    ema_fir_wmma<<<grid, 32 * WAVES_PER_BLOCK, 0, stream>>>(x, out);
    fix_head<<<(DDIM + 255) / 256, 256, 0, stream>>>(x, out);
}